// LieAugmenter_58566174048976
// MI455X (gfx1250) — compile-verified
//
#include <hip/hip_runtime.h>

typedef float v2f __attribute__((ext_vector_type(2)));
typedef float v8f __attribute__((ext_vector_type(8)));

#define NB     16384
#define DD     64
#define NG     3
#define NH     64
#define WPB    4                 // waves per block
#define BLOCK  (WPB * 32)
#define GROUPS (NB / 16)         // 1024 row-groups of 16
#define GRID   (GROUPS / WPB)    // 256 blocks
#define GSTR   66                // padded stride for transposed B-operand tables
#define TSTR   68                // padded stride for per-wave term tile
#define APPS   4                 // 2^s applications (s = 2)
#define TERMS  14                // Taylor terms per application
#define SCALE  0.25f             // 1 / 2^s

static __device__ __forceinline__ v8f wmma4(v2f a, v2f b, v8f c) {
  // D = A(16x4 f32) * B(4x16 f32) + C(16x16 f32)
  return __builtin_amdgcn_wmma_f32_16x16x4_f32(
      /*neg_a=*/false, a, /*neg_b=*/false, b,
      /*c_mod=*/(short)0, c, /*reuse_a=*/false, /*reuse_b=*/false);
}

__global__ __launch_bounds__(BLOCK) void lie_expm_mlp(
    const float* __restrict__ x,       // [B, 64]
    const float* __restrict__ alphas,  // [B, 3]
    const float* __restrict__ gen,     // [3, 64, 64]
    const float* __restrict__ w1,      // [64, 64]
    const float* __restrict__ b1,      // [64]
    const float* __restrict__ w2,      // [64]
    const float* __restrict__ b2,      // [1]
    float* __restrict__ out)           // [B]
{
  // Transposed so B-operand pairs (rows k, k+1 of a fixed column) are contiguous.
  __shared__ float GT[NG][DD][GSTR];   // GT[g][col][row] = gen[g][row][col]
  __shared__ float W1T[NH][GSTR];      // W1T[h][d]       = w1[d][h]
  __shared__ float Tb[WPB][16][TSTR];  // per-wave current term vector tile
  __shared__ float w2s[NH], b1s[NH];

  const int tid = threadIdx.x;
  for (int i = tid; i < NG * DD * DD; i += BLOCK) {
    int g = i / (DD * DD), rc = i % (DD * DD);
    GT[g][rc % DD][rc / DD] = gen[i];
  }
  for (int i = tid; i < DD * NH; i += BLOCK)
    W1T[i % NH][i / NH] = w1[i];
  if (tid < NH) { w2s[tid] = w2[tid]; b1s[tid] = b1[tid]; }
  __syncthreads();

  const int wave = tid >> 5, lane = tid & 31;
  const int hi = lane >> 4;            // lane-half selects K-pair / row-half
  const int ln = lane & 15;
  const int group = blockIdx.x * WPB + wave;   // 0..1023, exact cover
  const int rowbase = group * 16;
  float (*T)[TSTR] = Tb[wave];

  // Load this group's 16 rows of x into the term tile.
  for (int i = lane; i < 16 * DD; i += 32)
    T[i >> 6][i & 63] = x[rowbase * DD + i];

  // Per-lane alpha scalars for the 8 rows this lane owns in C-layout.
  float al[NG][8];
#pragma unroll
  for (int v = 0; v < 8; ++v)
#pragma unroll
    for (int g = 0; g < NG; ++g)
      al[g][v] = alphas[(rowbase + v + 8 * hi) * NG + g];

  asm volatile("" ::: "memory");   // keep LDS fills before first A-operand reads

  v8f sum[4];
#pragma unroll 1
  for (int app = 0; app < APPS; ++app) {
    // sum = current vector (term 0), loaded in C layout
#pragma unroll
    for (int n = 0; n < 4; ++n)
#pragma unroll
      for (int v = 0; v < 8; ++v)
        sum[n][v] = T[v + 8 * hi][n * 16 + ln];

#pragma unroll 1
    for (int k = 1; k <= TERMS; ++k) {
      v8f r0[4] = {}, r1[4] = {}, r2[4] = {};
#pragma unroll 4
      for (int kb = 0; kb < 16; ++kb) {
        const float* ap = &T[ln][4 * kb + 2 * hi];
        v2f a; a.x = ap[0]; a.y = ap[1];
#pragma unroll
        for (int n = 0; n < 4; ++n) {
          const int col = n * 16 + ln, ro = 4 * kb + 2 * hi;
          v2f bg0, bg1, bg2;
          bg0.x = GT[0][col][ro]; bg0.y = GT[0][col][ro + 1];
          bg1.x = GT[1][col][ro]; bg1.y = GT[1][col][ro + 1];
          bg2.x = GT[2][col][ro]; bg2.y = GT[2][col][ro + 1];
          r0[n] = wmma4(a, bg0, r0[n]);
          r1[n] = wmma4(a, bg1, r1[n]);
          r2[n] = wmma4(a, bg2, r2[n]);
        }
      }
      const float inv = SCALE / (float)k;
      asm volatile("" ::: "memory");   // all A-operand reads before term overwrite
#pragma unroll
      for (int n = 0; n < 4; ++n)
#pragma unroll
        for (int v = 0; v < 8; ++v) {
          float nt = (al[0][v] * r0[n][v] + al[1][v] * r1[n][v] +
                      al[2][v] * r2[n][v]) * inv;
          sum[n][v] += nt;
          T[v + 8 * hi][n * 16 + ln] = nt;   // new term for next k
        }
      asm volatile("" ::: "memory");
    }
    // Vector after this application becomes input of the next one.
#pragma unroll
    for (int n = 0; n < 4; ++n)
#pragma unroll
      for (int v = 0; v < 8; ++v)
        T[v + 8 * hi][n * 16 + ln] = sum[n][v];
    asm volatile("" ::: "memory");
  }

  // h = relu(xa @ w1 + b1): xa sits in T, bias preloaded into accumulators.
  v8f hc[4];
#pragma unroll
  for (int n = 0; n < 4; ++n) {
    const float bb = b1s[n * 16 + ln];
#pragma unroll
    for (int v = 0; v < 8; ++v) hc[n][v] = bb;
  }
#pragma unroll 4
  for (int kb = 0; kb < 16; ++kb) {
    const float* ap = &T[ln][4 * kb + 2 * hi];
    v2f a; a.x = ap[0]; a.y = ap[1];
#pragma unroll
    for (int n = 0; n < 4; ++n) {
      const int col = n * 16 + ln, ro = 4 * kb + 2 * hi;
      v2f b; b.x = W1T[col][ro]; b.y = W1T[col][ro + 1];
      hc[n] = wmma4(a, b, hc[n]);
    }
  }

  // out = relu(h) @ w2 + b2 : 16-lane butterfly reduction over columns.
  const float b2v = b2[0];
#pragma unroll
  for (int v = 0; v < 8; ++v) {
    float p = 0.f;
#pragma unroll
    for (int n = 0; n < 4; ++n)
      p += fmaxf(hc[n][v], 0.f) * w2s[n * 16 + ln];
    p += __shfl_xor(p, 1, 32);
    p += __shfl_xor(p, 2, 32);
    p += __shfl_xor(p, 4, 32);
    p += __shfl_xor(p, 8, 32);
    if (ln == 0) out[rowbase + v + 8 * hi] = p + b2v;
  }
}

extern "C" void kernel_launch(void* const* d_in, const int* in_sizes, int n_in,
                              void* d_out, int out_size, void* d_ws, size_t ws_size,
                              hipStream_t stream) {
  (void)in_sizes; (void)n_in; (void)out_size; (void)d_ws; (void)ws_size;
  const float* x      = (const float*)d_in[0];
  const float* alphas = (const float*)d_in[1];
  const float* gen    = (const float*)d_in[2];
  const float* w1     = (const float*)d_in[3];
  const float* b1     = (const float*)d_in[4];
  const float* w2     = (const float*)d_in[5];
  const float* b2     = (const float*)d_in[6];
  lie_expm_mlp<<<GRID, BLOCK, 0, stream>>>(x, alphas, gen, w1, b1, w2, b2,
                                           (float*)d_out);
}